// RelativePositionEncoding_11055245820591
// MI455X (gfx1250) — compile-verified
//
#include <hip/hip_runtime.h>

// CDNA5 / gfx1250, wave32.
typedef __attribute__((ext_vector_type(2))) float v2f;
typedef __attribute__((ext_vector_type(4))) float v4f;
typedef __attribute__((ext_vector_type(8))) float v8f;

#define CZ    128      // output channels
#define NCID  64       // 54 used combos, padded to 4 WMMA row-tiles
#define RMAXJ 2048     // cap on r for the LDS cid table (reference r = 1536)

// The only 10 rows of W that can ever be selected by the (buggy-but-faithful)
// one-hot logic: rel_pos rows {32,33,65}, rel_tok rows {98,99,131},
// same_entity row {132}, rel_chain rows {133,134,138}.
__device__ const int ROWMAP[10] = {32, 33, 65, 98, 99, 131, 132, 133, 134, 138};

// Selector matrix S[cid][k] (0/1), K padded to 12.
// cid = ((res_sel*3 + tok_sel)*2 + se)*3 + chain_sel
__device__ __forceinline__ float selS(int cid, int k) {
  int chain = cid % 3;
  int se    = (cid / 3) % 2;
  int tok   = (cid / 6) % 3;
  int res   = cid / 18;
  int v = 0;
  if (k < 3)       v = (k == res);
  else if (k < 6)  v = ((k - 3) == tok);
  else if (k == 6) v = se;
  else if (k < 10) v = ((k - 7) == chain);
  return (float)v;   // k = 10,11 -> 0 (K padding)
}

__global__ __launch_bounds__(256) void relpos_kernel(
    const int*  __restrict__ asym_id,
    const int*  __restrict__ residue_index,
    const int*  __restrict__ entity_id,
    const int*  __restrict__ token_index,
    const int*  __restrict__ sym_id,
    const float* __restrict__ W,      // [139,128] row-major
    float*      __restrict__ out,     // [b,r,r,128]
    int r)
{
  __shared__ __align__(16) float combos[NCID * CZ];  // 32 KB: combo table
  __shared__ int cidj[RMAXJ];                        // 8 KB: combo id per j

  const int tid  = threadIdx.x;
  const int wave = tid >> 5;      // 0..7, owns channel slice [16*wave, 16*wave+16)
  const int lane = tid & 31;
  const int n    = lane & 15;     // N / M-within-tile index
  const int half = lane >> 4;     // 0: lanes 0-15, 1: lanes 16-31
  const int cbase = wave * 16;

  const int batch = blockIdx.x / r;
  const int i     = blockIdx.x % r;

  // ---------------- Phase 1: combo table via V_WMMA_F32_16X16X4_F32 --------
  // combo[54][128] = S[54][10(+2 pad)] x Wsub[10(+2)][128], exact f32.
  // B operand (4x16, K x N): v[0] holds K = 4*kk + 2*half, v[1] holds K+1.
  v2f B0, B1, B2;
  {
    int k;
    k = 0 + 2 * half; B0.x = W[ROWMAP[k] * CZ + cbase + n];
    k = 1 + 2 * half; B0.y = W[ROWMAP[k] * CZ + cbase + n];
    k = 4 + 2 * half; B1.x = W[ROWMAP[k] * CZ + cbase + n];
    k = 5 + 2 * half; B1.y = W[ROWMAP[k] * CZ + cbase + n];
    k = 8 + 2 * half; B2.x = (k < 10) ? W[ROWMAP[k] * CZ + cbase + n] : 0.0f;
    k = 9 + 2 * half; B2.y = (k < 10) ? W[ROWMAP[k] * CZ + cbase + n] : 0.0f;
  }

  #pragma unroll
  for (int t = 0; t < 4; ++t) {
    // A operand (16x4, M x K): lane m = lane%16 holds row M = m;
    // v[0]: K = 4*kk + 2*half, v[1]: K = 4*kk + 2*half + 1.
    const int cidA = t * 16 + n;
    v2f A0, A1, A2;
    A0.x = selS(cidA, 0 + 2 * half); A0.y = selS(cidA, 1 + 2 * half);
    A1.x = selS(cidA, 4 + 2 * half); A1.y = selS(cidA, 5 + 2 * half);
    A2.x = selS(cidA, 8 + 2 * half); A2.y = selS(cidA, 9 + 2 * half);

    v8f acc = {};
    acc = __builtin_amdgcn_wmma_f32_16x16x4_f32(false, A0, false, B0,
                                                (short)0, acc, false, false);
    acc = __builtin_amdgcn_wmma_f32_16x16x4_f32(false, A1, false, B1,
                                                (short)0, acc, false, false);
    acc = __builtin_amdgcn_wmma_f32_16x16x4_f32(false, A2, false, B2,
                                                (short)0, acc, false, false);

    // D (16x16 f32): VGPR v, lanes 0-15: M=v, N=lane; lanes 16-31: M=v+8.
    #pragma unroll
    for (int v = 0; v < 8; ++v) {
      const int M = v + 8 * half;
      combos[(t * 16 + M) * CZ + cbase + n] = acc[v];
    }
  }

  // ---------------- Phase 2: classify every j into a combo id --------------
  const int* __restrict__ aB = asym_id       + (size_t)batch * r;
  const int* __restrict__ rB = residue_index + (size_t)batch * r;
  const int* __restrict__ eB = entity_id     + (size_t)batch * r;
  const int* __restrict__ sB = sym_id        + (size_t)batch * r;

  const int asym_i = aB[i];
  const int res_i  = rB[i];
  const int ent_i  = eB[i];
  const int sym_i  = sB[i];
  const int tok_i  = token_index[i];

  for (int j = tid; j < r; j += 256) {
    const int same_chain = (asym_i == aB[j]);
    const int same_res   = (res_i  == rB[j]);
    const int se         = (ent_i  == eB[j]);
    // faithful bug: residue_index[i] compared against entity_id[j]
    const int res_sel   = same_chain ? ((res_i == eB[j]) ? 1 : 0) : 2;
    const int tok_sel   = (same_chain & same_res)
                              ? ((tok_i == token_index[j]) ? 1 : 0) : 2;
    const int chain_sel = same_chain ? ((sym_i == sB[j]) ? 1 : 0) : 2;
    cidj[j] = ((res_sel * 3 + tok_sel) * 2 + se) * 3 + chain_sel;
  }

  __syncthreads();

  // ---------------- Phase 3: bandwidth-bound streaming store ----------------
  // One output row = 128 f32 = 512 B = one wave-wide global_store_b128.
  // Output (1.21 GB) is written once and never re-read: use non-temporal
  // stores so the stream does not thrash the 192 MB L2.
  float* __restrict__ outrow = out + (size_t)blockIdx.x * r * CZ;
  const int coff = lane * 4;     // lane covers channels [4*lane, 4*lane+4)

  #pragma unroll 4
  for (int j = wave; j < r; j += 8) {
    const int cid = cidj[j];
    const v4f v = *reinterpret_cast<const v4f*>(&combos[cid * CZ + coff]);
    __builtin_nontemporal_store(
        v, reinterpret_cast<v4f*>(&outrow[(size_t)j * CZ + coff]));
  }
}

extern "C" void kernel_launch(void* const* d_in, const int* in_sizes, int n_in,
                              void* d_out, int out_size, void* d_ws, size_t ws_size,
                              hipStream_t stream) {
  // setup_inputs order: aatype, asym_id, residue_index, entity_id,
  //                     token_index, sym_id, W
  const int*   asym = (const int*)d_in[1];
  const int*   resi = (const int*)d_in[2];
  const int*   ent  = (const int*)d_in[3];
  const int*   tok  = (const int*)d_in[4];
  const int*   sym  = (const int*)d_in[5];
  const float* W    = (const float*)d_in[6];
  float*       out  = (float*)d_out;

  const int r = in_sizes[4];             // token_index has shape [r]
  const int b = in_sizes[1] / r;         // asym_id has shape [b*r]

  relpos_kernel<<<dim3(b * r), dim3(256), 0, stream>>>(
      asym, resi, ent, tok, sym, W, out, r);
}